// Speech_LLM_Zipformer_31825707663837
// MI455X (gfx1250) — compile-verified
//
#include <hip/hip_runtime.h>
#include <hip/hip_bf16.h>
#include <limits.h>

// Problem constants (match reference setup_inputs)
#define BB 8
#define TT 512
#define SS 1024
#define HH 2048
#define LL (TT + SS - 1)   // 1535
#define START_TEXT   1
#define END_TEXT     2
#define START_SPEECH 3
#define END_SPEECH   4
#define DEF_SPEECH   5

// ---------------- CDNA5 async global<->LDS helpers (gfx1250) ----------------
// VGPR holds workgroup-relative LDS byte offset.
__device__ __forceinline__ void async_load16(unsigned lds_off, const float* gsrc) {
    asm volatile("global_load_async_to_lds_b128 %0, %1, off"
                 :: "v"(lds_off), "v"(gsrc) : "memory");
}
__device__ __forceinline__ void async_store16(float* gdst, unsigned lds_off) {
    asm volatile("global_store_async_from_lds_b128 %0, %1, off"
                 :: "v"(gdst), "v"(lds_off) : "memory");
}
__device__ __forceinline__ void wait_async0() {
    asm volatile("s_wait_asynccnt 0" ::: "memory");
}

// ---------------- Kernel 1: per-batch metadata ----------------
// meta[b*6+0]=n (valid length), +1=p (DEFAULT_SPEECH pos, 0 if absent),
// +2..5 = first index of START_TEXT, END_TEXT, START_SPEECH, END_SPEECH
//         (INT_MAX sentinel if absent -> never matches a text index < T)
__global__ void meta_kernel(const int* __restrict__ ids, int* __restrict__ meta) {
    __shared__ int sm[6];
    const int b = blockIdx.x;
    const int t = threadIdx.x;
    if (t < 6) sm[t] = INT_MAX;
    __syncthreads();
    for (int pos = t; pos < TT; pos += blockDim.x) {
        int id = ids[b * TT + pos];
        if      (id == 0)            atomicMin(&sm[0], pos);  // padding start -> n
        else if (id == DEF_SPEECH)   atomicMin(&sm[1], pos);
        else if (id == START_TEXT)   atomicMin(&sm[2], pos);
        else if (id == END_TEXT)     atomicMin(&sm[3], pos);
        else if (id == START_SPEECH) atomicMin(&sm[4], pos);
        else if (id == END_SPEECH)   atomicMin(&sm[5], pos);
    }
    __syncthreads();
    if (t < 6) {
        int v = sm[t];
        if (t == 0)      v = (v == INT_MAX) ? TT : v;  // no padding -> n = T
        else if (t == 1) v = (v == INT_MAX) ? 0  : v;  // argmax of empty mask -> 0
        meta[b * 6 + t] = v;
    }
}

// ---------------- Kernel 2: merged-row producer ----------------
// One 256-thread block per output row (b,l). Row copy = 2 x 128b async
// transfers per lane through LDS (CDNA5 async data-mover path).
__global__ void merge_kernel(const float* __restrict__ sp,     // [B,S,H]
                             const int*   __restrict__ slen,   // [B]
                             const float* __restrict__ emb,    // [B,T,H]
                             const int*   __restrict__ labels, // [B,T]
                             const float* __restrict__ pw,     // [4,H]
                             const int*   __restrict__ meta,   // [B,6]
                             float*       __restrict__ out) {
    __shared__ __align__(16) float ldsbuf[2 * 256 * 4];  // 8 KB
    const int bl  = blockIdx.x;
    const int b   = bl / LL;
    const int l   = bl - b * LL;
    const int tid = threadIdx.x;

    const int n  = meta[b * 6 + 0];
    const int p  = meta[b * 6 + 1];
    const int f0 = meta[b * 6 + 2];
    const int f1 = meta[b * 6 + 3];
    const int f2 = meta[b * 6 + 4];
    const int f3 = meta[b * 6 + 5];
    const int s  = slen[b];

    const int out_len = n + s - 1;
    const int k = l - (LL - out_len);   // position inside merged (unpadded) seq

    float* dst     = out + (size_t)bl * HH;
    float* maskout = out + (size_t)BB * LL * HH;
    float* lblout  = maskout + (size_t)BB * LL;

    if (k < 0 || k >= out_len) {
        // invalid (left padding) -> zeros, mask 0, label IGNORE
        const float4 z = make_float4(0.f, 0.f, 0.f, 0.f);
        *(float4*)(dst + tid * 4)        = z;
        *(float4*)(dst + 1024 + tid * 4) = z;
        if (tid == 0) { maskout[bl] = 0.f; lblout[bl] = -100.f; }
        return;
    }

    const float* src;
    float lbl;
    if (k >= p && k < p + s) {
        // speech frame
        src = sp + ((size_t)b * SS + (size_t)(k - p)) * HH;
        lbl = -100.f;
    } else {
        int ti = (k < p) ? k : (k - s + 1);
        ti = min(max(ti, 0), TT - 1);
        if      (ti == f0) src = pw + 0 * HH;
        else if (ti == f1) src = pw + 1 * HH;
        else if (ti == f2) src = pw + 2 * HH;
        else if (ti == f3) src = pw + 3 * HH;
        else               src = emb + ((size_t)b * TT + (size_t)ti) * HH;
        lbl = (float)labels[b * TT + ti];
    }

    // 8 KB row copy via async LDS data mover: two 16B chunks per lane,
    // distinct LDS slots so the second load never races the first store.
    const unsigned base = (unsigned)(size_t)(void*)ldsbuf;  // LDS byte offset
    const unsigned o0 = base + (unsigned)(tid * 16);
    const unsigned o1 = base + 4096u + (unsigned)(tid * 16);
    async_load16(o0, src + tid * 4);
    async_load16(o1, src + 1024 + tid * 4);
    wait_async0();
    async_store16(dst + tid * 4,        o0);
    async_store16(dst + 1024 + tid * 4, o1);
    // S_ENDPGM performs an implicit wait-idle, draining the async stores.

    if (tid == 0) { maskout[bl] = 1.f; lblout[bl] = lbl; }
}

extern "C" void kernel_launch(void* const* d_in, const int* in_sizes, int n_in,
                              void* d_out, int out_size, void* d_ws, size_t ws_size,
                              hipStream_t stream) {
    const float* sp     = (const float*)d_in[0];  // speech_features [B,S,H]
    const int*   slen   = (const int*)  d_in[1];  // speech_feature_lens [B]
    const float* emb    = (const float*)d_in[2];  // inputs_embeds [B,T,H]
    const int*   ids    = (const int*)  d_in[3];  // input_ids [B,T]
    // d_in[4] = attention_mask (bool) -- derived from input_ids instead
    const int*   labels = (const int*)  d_in[5];  // labels [B,T]
    const float* pw     = (const float*)d_in[6];  // prompt_weight [4,H]

    int*   meta = (int*)d_ws;       // B*6 ints of scratch
    float* out  = (float*)d_out;    // [B*L*H] emb | [B*L] mask | [B*L] labels

    meta_kernel<<<BB, 256, 0, stream>>>(ids, meta);
    merge_kernel<<<BB * LL, 256, 0, stream>>>(sp, slen, emb, labels, pw, meta, out);
}